// GCNNet_58205396795401
// MI455X (gfx1250) — compile-verified
//
#include <hip/hip_runtime.h>

typedef __attribute__((ext_vector_type(2))) float v2f;
typedef __attribute__((ext_vector_type(8))) float v8f;

#define DD 128   // feature dim

// ---------------------------------------------------------------------------
// Copy x into JK output slab column-block 0. out row stride = (L+1)*128.
// ---------------------------------------------------------------------------
__global__ __launch_bounds__(256) void copy_x_kernel(const float* __restrict__ x,
                                                     float* __restrict__ out,
                                                     int total, int out_stride) {
    int i = blockIdx.x * blockDim.x + threadIdx.x;
    if (i >= total) return;
    int n = i >> 7;
    int c = i & (DD - 1);
    out[(size_t)n * out_stride + c] = x[i];
}

// ---------------------------------------------------------------------------
// Degree with self loop: deg[n] = 1 + #edges with dst==n, then dinv = rsqrt.
// ---------------------------------------------------------------------------
__global__ __launch_bounds__(256) void deg_init_kernel(float* __restrict__ deg, int n) {
    int i = blockIdx.x * blockDim.x + threadIdx.x;
    if (i < n) deg[i] = 1.0f;   // self loop
}

__global__ __launch_bounds__(256) void deg_count_kernel(const int* __restrict__ ei,
                                                        float* __restrict__ deg, int e) {
    int i = blockIdx.x * blockDim.x + threadIdx.x;
    if (i < e) atomicAdd(&deg[ei[e + i]], 1.0f);   // dst row of edge_index
}

__global__ __launch_bounds__(256) void deg_rsqrt_kernel(float* __restrict__ deg, int n) {
    int i = blockIdx.x * blockDim.x + threadIdx.x;
    if (i < n) deg[i] = __frsqrt_rn(deg[i]);       // deg >= 1 always
}

// ---------------------------------------------------------------------------
// h = in @ W  using V_WMMA_F32_16X16X4_F32 (fp32-exact).
//
// W is staged into LDS in *fragment order*: Wlds2[k/2][col] = (W[k][col],
// W[k+1][col]) for even k. A lane's B fragment (rows kb+koff, kb+koff+1 at
// one column) is then a single contiguous ds_load_b64 into an even-aligned
// VGPR pair -> no v_dual_mov re-pairing in the WMMA inner loop.
//
// 8 waves / block, each wave owns a 16-row slab and all 8 column tiles.
// WMMA f32 16x16x4 lane layout (ISA 7.12.2):
//   A: lane l -> row l&15, K = 2*(l>>4) + {0,1}  (contiguous float2)
//   B: lane l -> col l&15, K = 2*(l>>4) + {0,1}
//   C/D: VGPR g -> row g + 8*(l>>4), col l&15
// ---------------------------------------------------------------------------
__global__ __launch_bounds__(256) void gemm_wmma_kernel(const float* __restrict__ in,
                                                        const float* __restrict__ W,
                                                        float* __restrict__ h,
                                                        int nrows, int in_stride) {
    __shared__ v2f Wlds2[(DD / 2) * DD];     // 64 KB, [kpair][col]
    for (int idx = threadIdx.x; idx < (DD / 2) * DD; idx += blockDim.x) {
        int p   = idx >> 7;        // k/2
        int col = idx & (DD - 1);
        v2f w;
        w.x = W[(2 * p)     * DD + col];
        w.y = W[(2 * p + 1) * DD + col];
        Wlds2[idx] = w;
    }
    __syncthreads();

    const int wave = threadIdx.x >> 5;
    const int lane = threadIdx.x & 31;
    const int rowbase = (blockIdx.x * 8 + wave) * 16;
    if (rowbase >= nrows) return;

    const int m    = lane & 15;         // A row / B col / C col within tile
    const int koff = (lane >> 4) << 1;  // 0 for lanes 0-15, 2 for lanes 16-31

    int arow_idx = rowbase + m;
    if (arow_idx >= nrows) arow_idx = nrows - 1;     // clamp (result discarded)
    const float* arow = in + (size_t)arow_idx * in_stride;

    // Per-lane base into Wlds2: kpair advances by 2 per kb step of 4.
    const v2f* bptr = Wlds2 + (koff >> 1) * DD + m;

    v8f acc[8] = {};

    for (int kb = 0; kb < DD; kb += 4) {
        v2f a;
        a.x = arow[kb + koff];
        a.y = arow[kb + koff + 1];
        const v2f* brow = bptr + (kb >> 1) * DD;
#pragma unroll
        for (int ct = 0; ct < 8; ct++) {
            v2f b = brow[ct * 16];
            acc[ct] = __builtin_amdgcn_wmma_f32_16x16x4_f32(
                false, a, false, b, (short)0, acc[ct], false, false);
        }
    }

    const int rofs = (lane >> 4) * 8;
#pragma unroll
    for (int ct = 0; ct < 8; ct++) {
#pragma unroll
        for (int g = 0; g < 8; g++) {
            int r = rowbase + rofs + g;
            if (r < nrows) h[(size_t)r * DD + ct * 16 + m] = acc[ct][g];
        }
    }
}

// ---------------------------------------------------------------------------
// agg[v,:] = h[v,:] * dinv[v]^2   (self-loop term; also initializes agg)
// ---------------------------------------------------------------------------
__global__ __launch_bounds__(256) void agg_init_kernel(const float* __restrict__ h,
                                                       const float* __restrict__ dinv,
                                                       float* __restrict__ agg, int total) {
    int i = blockIdx.x * blockDim.x + threadIdx.x;
    if (i >= total) return;
    float d = dinv[i >> 7];
    agg[i] = h[i] * d * d;
}

// ---------------------------------------------------------------------------
// One wave per edge: agg[dst,:] += h[src,:] * dinv[src]*dinv[dst]
// Each lane handles 4 contiguous columns (float4 load, 4 f32 atomics).
// h and agg (25.6 MB each) are L2-resident (192 MB L2), so the random
// gather/scatter stays on-chip.
// ---------------------------------------------------------------------------
__global__ __launch_bounds__(256) void scatter_kernel(const int* __restrict__ ei,
                                                      const float* __restrict__ h,
                                                      const float* __restrict__ dinv,
                                                      float* __restrict__ agg, int e) {
    int edge = blockIdx.x * (blockDim.x >> 5) + (threadIdx.x >> 5);
    int lane = threadIdx.x & 31;
    if (edge >= e) return;
    int s = ei[edge];
    int d = ei[e + edge];
    float w = dinv[s] * dinv[d];
    float4 hv = reinterpret_cast<const float4*>(h + (size_t)s * DD)[lane];
    float* ap = agg + (size_t)d * DD + lane * 4;
    atomicAdd(ap + 0, hv.x * w);
    atomicAdd(ap + 1, hv.y * w);
    atomicAdd(ap + 2, hv.z * w);
    atomicAdd(ap + 3, hv.w * w);
}

// ---------------------------------------------------------------------------
// BatchNorm statistics: column sums & sum-of-squares over N rows.
// ---------------------------------------------------------------------------
__global__ __launch_bounds__(256) void zero_stats_kernel(float* __restrict__ stats) {
    stats[threadIdx.x] = 0.0f;   // 256 floats: sums[128] | sq[128]
}

__global__ __launch_bounds__(256) void bn_stats_kernel(const float* __restrict__ agg,
                                                       float* __restrict__ sums,
                                                       float* __restrict__ sq, int n) {
    const int c    = threadIdx.x & (DD - 1);
    const int half = threadIdx.x >> 7;       // 0 or 1
    float s = 0.0f, q = 0.0f;
    for (int r = blockIdx.x * 2 + half; r < n; r += gridDim.x * 2) {
        float v = agg[(size_t)r * DD + c];
        s += v;
        q += v * v;
    }
    __shared__ float sh[512];
    sh[threadIdx.x]       = s;
    sh[256 + threadIdx.x] = q;
    __syncthreads();
    if (threadIdx.x < DD) {
        s = sh[threadIdx.x] + sh[threadIdx.x + DD];
        q = sh[256 + threadIdx.x] + sh[256 + threadIdx.x + DD];
        atomicAdd(&sums[threadIdx.x], s);
        atomicAdd(&sq[threadIdx.x], q);
    }
}

// scale = gamma * rsqrt(var+eps); shift = beta - mu*scale
// (GCNConv bias cancels exactly inside BatchNorm -> omitted.)
__global__ __launch_bounds__(128) void bn_finalize_kernel(const float* __restrict__ sums,
                                                          const float* __restrict__ sq,
                                                          const float* __restrict__ gamma,
                                                          const float* __restrict__ beta,
                                                          float* __restrict__ ss,
                                                          int n, float eps) {
    int c = threadIdx.x;
    float inv_n = 1.0f / (float)n;
    float mu  = sums[c] * inv_n;
    float var = sq[c] * inv_n - mu * mu;
    float rs  = __frsqrt_rn(var + eps);
    float sc  = gamma[c] * rs;
    ss[c]       = sc;
    ss[DD + c]  = beta[c] - mu * sc;
}

// o = relu(agg*scale + shift), written into the JK slab for layer l+1.
__global__ __launch_bounds__(256) void bn_apply_kernel(const float* __restrict__ agg,
                                                       const float* __restrict__ ss,
                                                       float* __restrict__ out,
                                                       int total, int out_stride) {
    int i = blockIdx.x * blockDim.x + threadIdx.x;
    if (i >= total) return;
    int n = i >> 7;
    int c = i & (DD - 1);
    float v = fmaf(agg[i], ss[c], ss[DD + c]);
    out[(size_t)n * out_stride + c] = v > 0.0f ? v : 0.0f;
}

// ---------------------------------------------------------------------------
extern "C" void kernel_launch(void* const* d_in, const int* in_sizes, int n_in,
                              void* d_out, int out_size, void* d_ws, size_t ws_size,
                              hipStream_t stream) {
    const float* x      = (const float*)d_in[0];
    const int*   ei     = (const int*)  d_in[1];
    const float* Ws     = (const float*)d_in[2];
    const float* bs     = (const float*)d_in[3];  (void)bs;  // cancels in BN
    const float* gammas = (const float*)d_in[4];
    const float* betas  = (const float*)d_in[5];
    float* out = (float*)d_out;

    const int N = in_sizes[0] / DD;
    const int E = in_sizes[1] / 2;
    const int L = in_sizes[2] / (DD * DD);
    const int out_stride = (L + 1) * DD;

    // Workspace layout (floats): dinv[N] | sums[128] | sq[128] | ss[256] | h[N*128] | agg[N*128]
    float* f    = (float*)d_ws;
    float* dinv = f;
    float* sums = dinv + N;
    float* sq   = sums + DD;
    float* ss   = sq   + DD;
    float* h    = ss   + 2 * DD;
    float* agg  = h + (size_t)N * DD;

    const int T = 256;
    int nd_blocks = (N * DD + T - 1) / T;

    copy_x_kernel<<<nd_blocks, T, 0, stream>>>(x, out, N * DD, out_stride);
    deg_init_kernel<<<(N + T - 1) / T, T, 0, stream>>>(dinv, N);
    deg_count_kernel<<<(E + T - 1) / T, T, 0, stream>>>(ei, dinv, E);
    deg_rsqrt_kernel<<<(N + T - 1) / T, T, 0, stream>>>(dinv, N);

    const int rows16      = (N + 15) / 16;
    const int gemm_blocks = (rows16 + 7) / 8;
    const int scat_blocks = (E + 7) / 8;

    for (int l = 0; l < L; l++) {
        const float* Wl = Ws + (size_t)l * DD * DD;
        gemm_wmma_kernel<<<gemm_blocks, T, 0, stream>>>(out + l * DD, Wl, h, N, out_stride);
        agg_init_kernel<<<nd_blocks, T, 0, stream>>>(h, dinv, agg, N * DD);
        scatter_kernel<<<scat_blocks, T, 0, stream>>>(ei, h, dinv, agg, E);
        zero_stats_kernel<<<1, 256, 0, stream>>>(sums);
        bn_stats_kernel<<<256, T, 0, stream>>>(agg, sums, sq, N);
        bn_finalize_kernel<<<1, DD, 0, stream>>>(sums, sq, gammas + l * DD, betas + l * DD,
                                                 ss, N, 1e-5f);
        bn_apply_kernel<<<nd_blocks, T, 0, stream>>>(agg, ss, out + (l + 1) * DD,
                                                     N * DD, out_stride);
    }
}